// AsymmetricLossCustomPriorityFinalAll_18064632447146
// MI455X (gfx1250) — compile-verified
//
#include <hip/hip_runtime.h>

#define TPB  256
#define TOPK 10

typedef __attribute__((ext_vector_type(2))) float v2f;
typedef __attribute__((ext_vector_type(8))) float v8f;

// Per-element contribution base*w with mult==1.
// y is exactly 0.0f or 1.0f, so the focal pow collapses to a select:
//   y==1: base = log(max(p,eps)),    w = (1-p)
//   y==0: base = log(max(neg,eps)),  w = (1-neg)^4,  neg = min(1.05-p, 1)
__device__ __forceinline__ float asl_contrib(float xv, float yv) {
    float p   = 1.0f / (1.0f + __expf(-xv));
    float neg = fminf(1.05f - p, 1.0f);
    bool  pos = yv > 0.5f;
    float a   = pos ? p : neg;
    float l   = __logf(fmaxf(a, 1e-8f));
    float w;
    if (pos) {
        w = 1.0f - p;
    } else {
        float t  = 1.0f - neg;
        float t2 = t * t;
        w = t2 * t2;
    }
    return l * w;
}

__global__ __launch_bounds__(TPB) void asl_row_kernel(
    const float* __restrict__ x, const float* __restrict__ y,
    const int* __restrict__ comp, const int* __restrict__ recy,
    const int* __restrict__ dona, const int* __restrict__ wl_map,
    int C, int nc, int nr, int nd, float* __restrict__ partial)
{
    __shared__ float s_v[TPB];
    __shared__ int   s_g[TPB];
    __shared__ int   s_t[TPB];
    __shared__ float s_red[TPB];
    __shared__ int   s_top[TOPK];
    __shared__ float s_f[TOPK];
    __shared__ float s_delta[TOPK];
    __shared__ int   s_flags[3];
    __shared__ int   s_winner;
    __shared__ float s_rowsum;

    const int tid = threadIdx.x;
    const long long rowoff = (long long)blockIdx.x * (long long)C;

    if (tid < 3) s_flags[tid] = 0;
    __syncthreads();

    // ---- per-row whitelist presence flags (170 scattered gathers) ----
    const int ntot = nc + nr + nd;
    if (tid < ntot) {
        int g, ci;
        if (tid < nc)            { g = 0; ci = comp[tid]; }
        else if (tid < nc + nr)  { g = 1; ci = recy[tid - nc]; }
        else                     { g = 2; ci = dona[tid - nc - nr]; }
        if (y[rowoff + ci] > 0.5f) atomicOr(&s_flags[g], 1);
    }

    // ---- single streaming pass: partial sum + register-resident top-10 ----
    float tv[TOPK]; int ti[TOPK];
    #pragma unroll
    for (int k = 0; k < TOPK; ++k) { tv[k] = -__builtin_inff(); ti[k] = 0x7fffffff; }

    float local = 0.0f;
    for (int i = tid; i < C; i += TPB) {
        float xv = x[rowoff + i];
        float yv = y[rowoff + i];
        local += asl_contrib(xv, yv);
        // stable descending insertion: strict > keeps earlier (smaller) index on ties
        if (xv > tv[TOPK - 1]) {
            int j = TOPK - 1;
            while (j > 0 && xv > tv[j - 1]) { tv[j] = tv[j - 1]; ti[j] = ti[j - 1]; --j; }
            tv[j] = xv; ti[j] = i;
        }
    }

    // ---- workgroup sum of 256 partials via V_WMMA_F32_16X16X4_F32 ----
    // D = ones(16x4) x B(4x16) + C  =>  every D[m][n] = column-sum of B.
    // Chain 4 WMMAs over 4 chunks of 64 LDS partials (exact f32 adds),
    // then 4 shfl_xor fold the 16 column sums. All waves execute (EXEC all-1s).
    s_red[tid] = local;
    __syncthreads();
    {
        const int lane = tid & 31;
        v2f A; A.x = 1.0f; A.y = 1.0f;
        v8f acc = {};
        #pragma unroll
        for (int c4 = 0; c4 < 4; ++c4) {
            v2f Bv;
            Bv.x = s_red[c4 * 64 + lane];
            Bv.y = s_red[c4 * 64 + 32 + lane];
            acc = __builtin_amdgcn_wmma_f32_16x16x4_f32(
                false, A, false, Bv, (short)0, acc, false, false);
        }
        float s = acc[0];            // lane l holds colsum(l % 16)
        #pragma unroll
        for (int m = 1; m < 16; m <<= 1) s += __shfl_xor(s, m, 32);
        if (tid == 0) s_rowsum = s;  // lane 0 = sum over lanes 0..15 = total
    }

    // ---- 256-way merge of sorted per-thread lists -> global top-10 ----
    int head = 0;
    for (int r = 0; r < TOPK; ++r) {
        bool ok = head < TOPK;
        s_v[tid] = ok ? tv[head] : -__builtin_inff();
        s_g[tid] = ok ? ti[head] : 0x7fffffff;
        s_t[tid] = tid;
        __syncthreads();
        for (int str = TPB >> 1; str > 0; str >>= 1) {
            if (tid < str) {
                float v1 = s_v[tid], v2 = s_v[tid + str];
                int   g1 = s_g[tid], g2 = s_g[tid + str];
                if (v2 > v1 || (v2 == v1 && g2 < g1)) {
                    s_v[tid] = v2; s_g[tid] = g2; s_t[tid] = s_t[tid + str];
                }
            }
            __syncthreads();
        }
        if (tid == 0) { s_top[r] = s_g[0]; s_winner = s_t[0]; }
        __syncthreads();
        if (tid == s_winner) ++head;
        __syncthreads();
    }

    // ---- sequential multiplier scan over the 10 ranks (order matters) ----
    if (tid == 0) {
        int h1 = s_flags[0], h2 = s_flags[1], h3 = s_flags[2];
        bool only4 = !(h1 | h2 | h3);
        bool found = false;
        for (int r = 0; r < TOPK; ++r) {
            int  j      = s_top[r];
            int  wl     = wl_map[j];
            bool in_map = wl > 0;
            bool in_gt  = ((wl == 1) && h1) || ((wl == 2) && h2) ||
                          ((wl == 3) && h3) || ((wl == 4) && only4);
            float f = 1.0f;
            if (in_map && only4) f *= 0.5f;                 // ALPHA_OTHER
            if (in_map && !in_gt && !found) f *= 2.0f;      // ALPHA1
            s_f[r] = f;
            found = found || (in_map && in_gt);
        }
        if (!found)
            for (int r = 0; r < TOPK; ++r) s_f[r] *= 2.0f;  // extra ALPHA1 pass
    }
    __syncthreads();

    // ---- corrections: add (f-1)*base*w at the 10 top-k positions ----
    if (tid < TOPK) {
        int j = s_top[tid];
        float c = asl_contrib(x[rowoff + j], y[rowoff + j]);
        s_delta[tid] = (s_f[tid] - 1.0f) * c;
    }
    __syncthreads();

    if (tid == 0) {
        float tot = s_rowsum;
        for (int r = 0; r < TOPK; ++r) tot += s_delta[r];
        partial[blockIdx.x] = tot;
    }
}

__global__ __launch_bounds__(TPB) void asl_final_reduce(
    const float* __restrict__ partial, int n, float* __restrict__ out)
{
    __shared__ float s[TPB];
    float v = 0.0f;
    for (int i = threadIdx.x; i < n; i += TPB) v += partial[i];
    s[threadIdx.x] = v;
    __syncthreads();
    for (int str = TPB >> 1; str > 0; str >>= 1) {
        if (threadIdx.x < str) s[threadIdx.x] += s[threadIdx.x + str];
        __syncthreads();
    }
    if (threadIdx.x == 0) out[0] = -s[0];
}

extern "C" void kernel_launch(void* const* d_in, const int* in_sizes, int n_in,
                              void* d_out, int out_size, void* d_ws, size_t ws_size,
                              hipStream_t stream) {
    const float* x    = (const float*)d_in[0];
    const float* y    = (const float*)d_in[1];
    const int*   comp = (const int*)d_in[2];
    const int*   recy = (const int*)d_in[3];
    const int*   dona = (const int*)d_in[4];
    const int*   wl   = (const int*)d_in[5];

    const int C  = in_sizes[5];            // wl_map has C entries
    const int B  = in_sizes[0] / C;        // rows
    const int nc = in_sizes[2];
    const int nr = in_sizes[3];
    const int nd = in_sizes[4];

    float* partial = (float*)d_ws;         // B floats of scratch
    float* out     = (float*)d_out;

    asl_row_kernel<<<B, TPB, 0, stream>>>(x, y, comp, recy, dona, wl,
                                          C, nc, nr, nd, partial);
    asl_final_reduce<<<1, TPB, 0, stream>>>(partial, B, out);
}